// Renderer_27444841021604
// MI455X (gfx1250) — compile-verified
//
#include <hip/hip_runtime.h>
#include <math.h>

#define B_   4
#define V_   5000
#define F_   10000
#define H_   512
#define W_   512
#define UV_  512
#define HW_  (H_*W_)

// d_out section offsets (floats). B*HW = 1048576.
#define OFF_IMG   0
#define OFF_ALB   (3*B_*HW_)
#define OFF_ALPHA (6*B_*HW_)
#define OFF_MASK  (7*B_*HW_)
#define OFF_SHD   (8*B_*HW_)
#define OFF_GRID  (11*B_*HW_)
#define OFF_NRM   (13*B_*HW_)

typedef float v2f __attribute__((ext_vector_type(2)));
typedef float v8f __attribute__((ext_vector_type(8)));

// ---------------------------------------------------------------- utilities
__device__ __forceinline__ void cross3(const float a[3], const float b[3], float c[3]) {
    c[0] = a[1]*b[2] - a[2]*b[1];
    c[1] = a[2]*b[0] - a[0]*b[2];
    c[2] = a[0]*b[1] - a[1]*b[0];
}

__device__ __forceinline__ void accum_face(const float v0[3], const float v1[3], const float v2[3],
                                           float* acc, int base0, int base1, int base2) {
    float e0[3], e1[3], cr[3];
    // at vertex1: cross(v2 - v1, v0 - v1)
    for (int c = 0; c < 3; ++c) { e0[c] = v2[c] - v1[c]; e1[c] = v0[c] - v1[c]; }
    cross3(e0, e1, cr);
    atomicAdd(&acc[base1+0], cr[0]); atomicAdd(&acc[base1+1], cr[1]); atomicAdd(&acc[base1+2], cr[2]);
    // at vertex2: cross(v0 - v2, v1 - v2)
    for (int c = 0; c < 3; ++c) { e0[c] = v0[c] - v2[c]; e1[c] = v1[c] - v2[c]; }
    cross3(e0, e1, cr);
    atomicAdd(&acc[base2+0], cr[0]); atomicAdd(&acc[base2+1], cr[1]); atomicAdd(&acc[base2+2], cr[2]);
    // at vertex0: cross(v1 - v0, v2 - v0)
    for (int c = 0; c < 3; ++c) { e0[c] = v1[c] - v0[c]; e1[c] = v2[c] - v0[c]; }
    cross3(e0, e1, cr);
    atomicAdd(&acc[base0+0], cr[0]); atomicAdd(&acc[base0+1], cr[1]); atomicAdd(&acc[base0+2], cr[2]);
}

// ---------------------------------------------------------------- kernels
__global__ void k_zero(float* __restrict__ p, int n) {
    int i = blockIdx.x * blockDim.x + threadIdx.x;
    if (i < n) p[i] = 0.0f;
}

__global__ void k_face_scatter(const float* __restrict__ verts,
                               const float* __restrict__ tverts,
                               const int*   __restrict__ faces,
                               float* __restrict__ acc_n,
                               float* __restrict__ acc_tn) {
    int t = blockIdx.x * blockDim.x + threadIdx.x;
    if (t >= B_ * F_) return;
    int b = t / F_, f = t - b * F_;
    int i0 = faces[f*3+0], i1 = faces[f*3+1], i2 = faces[f*3+2];
    const float* vb = verts  + b * V_ * 3;
    const float* tb = tverts + b * V_ * 3;
    float v0[3], v1[3], v2[3], t0[3], t1[3], t2[3];
    #pragma unroll
    for (int c = 0; c < 3; ++c) {
        v0[c] = vb[i0*3+c]; v1[c] = vb[i1*3+c]; v2[c] = vb[i2*3+c];
        t0[c] = tb[i0*3+c]; t1[c] = tb[i1*3+c]; t2[c] = tb[i2*3+c];
    }
    // reference adds 10 to z of transformed verts before normal computation
    t0[2] += 10.0f; t1[2] += 10.0f; t2[2] += 10.0f;
    int b0 = (b*V_ + i0)*3, b1 = (b*V_ + i1)*3, b2 = (b*V_ + i2)*3;
    accum_face(v0, v1, v2, acc_n,  b0, b1, b2);
    accum_face(t0, t1, t2, acc_tn, b0, b1, b2);
}

__global__ void k_normalize(const float* __restrict__ acc_n, const float* __restrict__ acc_tn,
                            float* __restrict__ nrm_n, float* __restrict__ nrm_tn,
                            float* __restrict__ out_nrm) {
    int t = blockIdx.x * blockDim.x + threadIdx.x;
    if (t >= B_ * V_) return;
    {
        float x = acc_n[t*3+0], y = acc_n[t*3+1], z = acc_n[t*3+2];
        float inv = 1.0f / fmaxf(sqrtf(x*x + y*y + z*z), 1e-6f);
        float nx = x*inv, ny = y*inv, nz = z*inv;
        nrm_n[t*3+0] = nx; nrm_n[t*3+1] = ny; nrm_n[t*3+2] = nz;
        out_nrm[t*3+0] = nx; out_nrm[t*3+1] = ny; out_nrm[t*3+2] = nz;
    }
    {
        float x = acc_tn[t*3+0], y = acc_tn[t*3+1], z = acc_tn[t*3+2];
        float inv = 1.0f / fmaxf(sqrtf(x*x + y*y + z*z), 1e-6f);
        nrm_tn[t*3+0] = x*inv; nrm_tn[t*3+1] = y*inv; nrm_tn[t*3+2] = z*inv;
    }
}

// attributes[b][f][k][0:12] = [fuv(3), tfn(3), fv(3), fn(3)]
__global__ void k_attr(const float* __restrict__ verts,
                       const float* __restrict__ fuv,
                       const int*   __restrict__ faces,
                       const float* __restrict__ nrm_n,
                       const float* __restrict__ nrm_tn,
                       float* __restrict__ attrb) {
    int t = blockIdx.x * blockDim.x + threadIdx.x;      // t = (b*F + f)*3 + k
    if (t >= B_ * F_ * 3) return;
    int k  = t % 3;
    int bf = t / 3;
    int f  = bf % F_;
    int b  = bf / F_;
    int vid = faces[f*3 + k];
    int vbase = (b*V_ + vid)*3;
    float* dst = attrb + (long)t * 12;
    #pragma unroll
    for (int c = 0; c < 3; ++c) {
        dst[0 + c] = fuv[(f*3 + k)*3 + c];
        dst[3 + c] = nrm_tn[vbase + c];
        dst[6 + c] = verts[vbase + c];
        dst[9 + c] = nrm_n[vbase + c];
    }
}

// fused per-pixel kernel: rasterize-gather + bilinear albedo + WMMA SH shading
__global__ __launch_bounds__(256) void k_render(const float* __restrict__ albedos,
                                                const float* __restrict__ lights,
                                                const float* __restrict__ bary,
                                                const int*   __restrict__ p2f,
                                                const float* __restrict__ attrb,
                                                float* __restrict__ out) {
    __shared__ float sh_lds[8][32*12];    // [wave][pixel-in-wave][k]  (SH basis, K padded 9->12)
    __shared__ float shade_lds[8][32*3];  // [wave][pixel-in-wave][channel]

    const int tid  = threadIdx.x;
    const int wv   = tid >> 5;
    const int lane = tid & 31;
    const int p    = blockIdx.x * 256 + tid;          // global pixel id, exact cover of B*H*W
    const int b    = p >> 18;                         // / (H*W) = 2^18
    const int r    = p & (HW_ - 1);
    const int y    = r >> 9;
    const int x    = r & (W_ - 1);

    const int   idx = p2f[p];
    const float w0 = bary[p*3+0], w1 = bary[p*3+1], w2 = bary[p*3+2];

    float pv[12];
    float vis;
    if (idx >= 0) {
        const float* a = attrb + (long)idx * 36;
        #pragma unroll
        for (int d = 0; d < 12; ++d)
            pv[d] = w0*a[d] + w1*a[12+d] + w2*a[24+d];
        vis = 1.0f;
    } else {
        #pragma unroll
        for (int d = 0; d < 12; ++d) pv[d] = 0.0f;
        vis = 0.0f;
    }

    // grid / alpha / pos_mask
    out[OFF_GRID + p*2 + 0] = pv[0];
    out[OFF_GRID + p*2 + 1] = pv[1];
    out[OFF_ALPHA + p] = vis;
    out[OFF_MASK  + p] = (pv[5] < -0.05f) ? 1.0f : 0.0f;

    // bilinear albedo sample at (pv0, pv1), zero-padded borders (matches reference)
    float gx = (pv[0] + 1.0f) * 0.5f * (float)UV_ - 0.5f;
    float gy = (pv[1] + 1.0f) * 0.5f * (float)UV_ - 0.5f;
    float x0f = floorf(gx), y0f = floorf(gy);
    float wx1 = gx - x0f, wx0 = 1.0f - wx1;
    float wy1 = gy - y0f, wy0 = 1.0f - wy1;
    float alb[3] = {0.0f, 0.0f, 0.0f};
    const float* ab = albedos + (long)b * 3 * UV_ * UV_;
    #pragma unroll
    for (int cy = 0; cy < 2; ++cy) {
        float iyf = y0f + (float)cy;
        float wy  = cy ? wy1 : wy0;
        #pragma unroll
        for (int cx = 0; cx < 2; ++cx) {
            float ixf = x0f + (float)cx;
            float wgt = (cx ? wx1 : wx0) * wy;
            bool valid = (ixf >= 0.0f) && (ixf <= (float)(UV_-1)) &&
                         (iyf >= 0.0f) && (iyf <= (float)(UV_-1));
            int ix = (int)fminf(fmaxf(ixf, 0.0f), (float)(UV_-1));
            int iy = (int)fminf(fmaxf(iyf, 0.0f), (float)(UV_-1));
            float wv2 = valid ? wgt : 0.0f;
            int base = iy * UV_ + ix;
            alb[0] += wv2 * ab[0*UV_*UV_ + base];
            alb[1] += wv2 * ab[1*UV_*UV_ + base];
            alb[2] += wv2 * ab[2*UV_*UV_ + base];
        }
    }

    // SH basis (with CONST_FACTOR folded in), padded to K=12
    {
        float nx = pv[9], ny = pv[10], nz = pv[11];
        float sh[12];
        sh[0]  = 0.28209479177387814f;
        sh[1]  = 1.0233267079464885f * nx;
        sh[2]  = 1.0233267079464885f * ny;
        sh[3]  = 1.0233267079464885f * nz;
        sh[4]  = 0.8580855308097834f * nx*ny;
        sh[5]  = 0.8580855308097834f * nx*nz;
        sh[6]  = 0.8580855308097834f * ny*nz;
        sh[7]  = 0.4290427654048917f * (nx*nx - ny*ny);
        sh[8]  = 0.24770795610037571f * (3.0f*nz*nz - 1.0f);
        sh[9]  = 0.0f; sh[10] = 0.0f; sh[11] = 0.0f;
        #pragma unroll
        for (int k = 0; k < 12; ++k) sh_lds[wv][lane*12 + k] = sh[k];
    }
    __syncthreads();

    // A-matrix (16x4 f32 layout): row m = lane&15 (rows 0..2 = RGB of lights),
    // VGPR0 holds K = k0 (lanes 0-15) / k0+2 (lanes 16-31); VGPR1 = +1. K padded 9->12.
    const int m     = lane & 15;
    const int khalf = (lane < 16) ? 0 : 2;
    v2f Amat[3];
    #pragma unroll
    for (int kw = 0; kw < 3; ++kw) {
        int ka = kw*4 + khalf;
        float a0 = 0.0f, a1 = 0.0f;
        if (m < 3) {
            if (ka     < 9) a0 = lights[b*27 + ka*3 + m];        // lights[b, k, c=m]
            if (ka + 1 < 9) a1 = lights[b*27 + (ka+1)*3 + m];
        }
        Amat[kw].x = a0; Amat[kw].y = a1;
    }

    // Two 16-pixel groups per wave; D = L(3x12) x SH(12x16) via chained 16x16x4 f32 WMMA.
    #pragma unroll
    for (int g = 0; g < 2; ++g) {
        const float* sb = &sh_lds[wv][g*16*12];
        const int col = lane & 15;                     // B-matrix column = pixel in group
        v8f acc = {0.0f,0.0f,0.0f,0.0f,0.0f,0.0f,0.0f,0.0f};
        #pragma unroll
        for (int kw = 0; kw < 3; ++kw) {
            v2f Bm;
            Bm.x = sb[col*12 + kw*4 + khalf];          // K-striped like A: K0|K2, K1|K3
            Bm.y = sb[col*12 + kw*4 + khalf + 1];
            acc = __builtin_amdgcn_wmma_f32_16x16x4_f32(
                false, Amat[kw], false, Bm, (short)0, acc, false, false);
        }
        if (lane < 16) {                               // rows 0..2 of D live on lanes 0-15
            shade_lds[wv][(g*16 + lane)*3 + 0] = acc[0];
            shade_lds[wv][(g*16 + lane)*3 + 1] = acc[1];
            shade_lds[wv][(g*16 + lane)*3 + 2] = acc[2];
        }
    }
    __syncthreads();

    float s0 = shade_lds[wv][lane*3 + 0];
    float s1 = shade_lds[wv][lane*3 + 1];
    float s2 = shade_lds[wv][lane*3 + 2];

    // images / albedo / shading outputs, [B,3,H,W]
    int ipx = b*3*HW_ + y*W_ + x;
    out[OFF_IMG + ipx        ] = alb[0]*s0*vis;
    out[OFF_IMG + ipx +   HW_] = alb[1]*s1*vis;
    out[OFF_IMG + ipx + 2*HW_] = alb[2]*s2*vis;
    out[OFF_ALB + ipx        ] = alb[0];
    out[OFF_ALB + ipx +   HW_] = alb[1];
    out[OFF_ALB + ipx + 2*HW_] = alb[2];
    out[OFF_SHD + ipx        ] = s0;
    out[OFF_SHD + ipx +   HW_] = s1;
    out[OFF_SHD + ipx + 2*HW_] = s2;
}

// ---------------------------------------------------------------- launch
extern "C" void kernel_launch(void* const* d_in, const int* in_sizes, int n_in,
                              void* d_out, int out_size, void* d_ws, size_t ws_size,
                              hipStream_t stream) {
    const float* verts   = (const float*)d_in[0];   // [B,V,3]
    const float* tverts  = (const float*)d_in[1];   // [B,V,3]
    const float* albedos = (const float*)d_in[2];   // [B,3,UV,UV]
    const float* lights  = (const float*)d_in[3];   // [B,9,3]
    const float* fuv     = (const float*)d_in[4];   // [1,F,3,3]
    const float* bary    = (const float*)d_in[5];   // [B,H,W,1,3]
    const int*   faces   = (const int*)d_in[6];     // [1,F,3]
    const int*   p2f     = (const int*)d_in[7];     // [B,H,W,1]
    float* out = (float*)d_out;

    float* ws     = (float*)d_ws;
    float* acc_n  = ws;                       // B*V*3 = 60000
    float* acc_tn = acc_n  + B_*V_*3;         // 60000
    float* nrm_n  = acc_tn + B_*V_*3;         // 60000
    float* nrm_tn = nrm_n  + B_*V_*3;         // 60000
    float* attrb  = nrm_tn + B_*V_*3;         // B*F*3*12 = 1,440,000

    const int nAcc = 2 * B_ * V_ * 3;
    k_zero<<<(nAcc + 255)/256, 256, 0, stream>>>(acc_n, nAcc);

    k_face_scatter<<<(B_*F_ + 255)/256, 256, 0, stream>>>(verts, tverts, faces, acc_n, acc_tn);

    k_normalize<<<(B_*V_ + 255)/256, 256, 0, stream>>>(acc_n, acc_tn, nrm_n, nrm_tn,
                                                       out + OFF_NRM);

    k_attr<<<(B_*F_*3 + 255)/256, 256, 0, stream>>>(verts, fuv, faces, nrm_n, nrm_tn, attrb);

    k_render<<<(B_*HW_)/256, 256, 0, stream>>>(albedos, lights, bary, p2f, attrb, out);
}